// MyLoss_1159641170324
// MI455X (gfx1250) — compile-verified
//
#include <hip/hip_runtime.h>
#include <stdint.h>

#define THREADS 128
#define NBLK 1024

typedef float v2f __attribute__((ext_vector_type(2)));
typedef float v8f __attribute__((ext_vector_type(8)));

// ---- CDNA5 async global->LDS copies (ASYNCcnt-tracked), GVS addressing ----
__device__ __forceinline__ void async_b128(uint32_t lds, uint64_t saddr, uint32_t voff) {
  asm volatile("global_load_async_to_lds_b128 %0, %1, %2"
               :: "v"(lds), "v"(voff), "s"(saddr) : "memory");
}
__device__ __forceinline__ void async_b32(uint32_t lds, uint64_t saddr, uint32_t voff) {
  asm volatile("global_load_async_to_lds_b32 %0, %1, %2"
               :: "v"(lds), "v"(voff), "s"(saddr) : "memory");
}
__device__ __forceinline__ void wait_async0() {
  asm volatile("s_wait_asynccnt 0" ::: "memory");
}

// ---- f32 WMMA wave32 reduction: A(16x4)=partials, B(4x16)=ones -> row sums ----
// A layout: lane L(0..15) vgpr0=A[L,0], vgpr1=A[L,1]; lane L+16 vgpr0=A[L,2].
// With a.y=0, b=ones: D[m,n] = p[m] + p[m+16] for all n.
// Lane<16 holds D rows 0..7 in c[0..7]; lane>=16 rows 8..15.
__device__ __forceinline__ float wave_sum32(float x) {
  v2f a; a[0] = x;    a[1] = 0.0f;
  v2f b; b[0] = 1.0f; b[1] = 1.0f;
  v8f c = {};
  c = __builtin_amdgcn_wmma_f32_16x16x4_f32(false, a, false, b, (short)0, c,
                                            false, false);
  float s = c[0] + c[1] + c[2] + c[3] + c[4] + c[5] + c[6] + c[7];
  // lanes 0..15: sum_{m=0..7}(p[m]+p[m+16]); lanes 16..31: sum_{m=8..15}(...)
  return __shfl(s, 0, 32) + __shfl(s, 16, 32);
}

__global__ __launch_bounds__(THREADS) void spike_loss_pass1(
    const float* __restrict__ gin, const float* __restrict__ gtg,
    float* __restrict__ part, int nbody) {
  __shared__ __align__(16) float sIn[THREADS * 45];
  __shared__ __align__(16) float sTg[THREADS * 45];
  __shared__ float red[(THREADS / 32) * 3];

  constexpr int conn[14][2] = {
      {0,1},{1,2},{1,3},{2,4},{3,5},{4,6},{5,7},
      {1,8},{8,9},{8,10},{9,11},{10,12},{11,13},{12,14}};
  constexpr int sym[6][4] = {
      {2,4,3,5},{4,6,5,7},{9,11,10,12},{11,13,12,14},{1,2,1,3},{8,9,8,10}};

  const int tid = threadIdx.x;
  const uint64_t gin_u = (uint64_t)(uintptr_t)gin;
  const uint64_t gtg_u = (uint64_t)(uintptr_t)gtg;
  const uint32_t ldsIn0 = (uint32_t)(uintptr_t)&sIn[0];  // low 32 bits == LDS addr
  const uint32_t ldsTg0 = (uint32_t)(uintptr_t)&sTg[0];

  float a1 = 0.f, a2 = 0.f, a3 = 0.f;
  const int ntiles = (nbody + THREADS - 1) / THREADS;

  for (int tile = blockIdx.x; tile < ntiles; tile += gridDim.x) {
    const int base = tile * THREADS;
    const int nb = min(THREADS, nbody - base);
    const uint32_t gbyte = (uint32_t)base * 180u;  // 45 floats * 4B per body
    const int nflt = nb * 45, n4 = nflt >> 2, rem = nflt & 3;

    // coalesced async staging: 16B per lane per instruction, contiguous
    for (int i = tid; i < n4; i += THREADS) {
      const uint32_t o = (uint32_t)i * 16u;
      async_b128(ldsIn0 + o, gin_u, gbyte + o);
      async_b128(ldsTg0 + o, gtg_u, gbyte + o);
    }
    for (int j = tid; j < rem; j += THREADS) {
      const uint32_t o = (uint32_t)n4 * 16u + (uint32_t)j * 4u;
      async_b32(ldsIn0 + o, gin_u, gbyte + o);
      async_b32(ldsTg0 + o, gtg_u, gbyte + o);
    }
    wait_async0();      // each wave drains its own ASYNCcnt
    __syncthreads();    // then all waves' copies are visible

    if (tid < nb) {
      float vin[45], vtg[45];
      const float* xin = &sIn[tid * 45];
      const float* xtg = &sTg[tid * 45];
#pragma unroll
      for (int i = 0; i < 45; ++i) { vin[i] = xin[i]; vtg[i] = xtg[i]; }

      float p1 = 0.f;
#pragma unroll
      for (int i = 0; i < 45; ++i) p1 += __builtin_fabsf(vin[i] - vtg[i]);

      float p2 = 0.f;
#pragma unroll
      for (int l = 0; l < 14; ++l) {
        const int a = conn[l][0] * 3, b = conn[l][1] * 3;
        float li = 0.f, lt = 0.f;
#pragma unroll
        for (int cidx = 0; cidx < 3; ++cidx) {
          li += __builtin_fabsf(vin[a + cidx] - vin[b + cidx]);
          lt += __builtin_fabsf(vtg[a + cidx] - vtg[b + cidx]);
        }
        p2 += __builtin_fabsf(li - lt);
      }

      float p3 = 0.f;
#pragma unroll
      for (int s = 0; s < 6; ++s) {
        const int a = sym[s][0] * 3, b = sym[s][1] * 3;
        const int c0 = sym[s][2] * 3, d = sym[s][3] * 3;
        float lf = 0.f, rt = 0.f;
#pragma unroll
        for (int cidx = 0; cidx < 3; ++cidx) {
          lf += __builtin_fabsf(vin[a + cidx] - vin[b + cidx]);
          rt += __builtin_fabsf(vtg[c0 + cidx] - vtg[d + cidx]);
        }
        p3 += __builtin_fabsf(lf - rt);
      }
      a1 += p1; a2 += p2; a3 += p3;
    }
    __syncthreads();  // protect LDS before next tile's async writes
  }

  // wave-level reduction via f32 WMMA (EXEC all-ones here: no divergence)
  const float w1 = wave_sum32(a1);
  const float w2 = wave_sum32(a2);
  const float w3 = wave_sum32(a3);

  const int lane = tid & 31, wv = tid >> 5;
  if (lane == 0) { red[wv * 3 + 0] = w1; red[wv * 3 + 1] = w2; red[wv * 3 + 2] = w3; }
  __syncthreads();
  if (tid == 0) {
    float s1 = 0.f, s2 = 0.f, s3 = 0.f;
    for (int w = 0; w < THREADS / 32; ++w) {
      s1 += red[w * 3 + 0]; s2 += red[w * 3 + 1]; s3 += red[w * 3 + 2];
    }
    part[blockIdx.x * 3 + 0] = s1;
    part[blockIdx.x * 3 + 1] = s2;
    part[blockIdx.x * 3 + 2] = s3;
  }
}

__global__ __launch_bounds__(256) void spike_loss_pass2(
    const float* __restrict__ part, float* __restrict__ out, int nparts,
    float inv1, float inv23) {
  __shared__ float r1[256], r2[256], r3[256];
  const int tid = threadIdx.x;
  float s1 = 0.f, s2 = 0.f, s3 = 0.f;
  for (int i = tid; i < nparts; i += 256) {
    s1 += part[i * 3 + 0]; s2 += part[i * 3 + 1]; s3 += part[i * 3 + 2];
  }
  r1[tid] = s1; r2[tid] = s2; r3[tid] = s3;
  __syncthreads();
  for (int off = 128; off > 0; off >>= 1) {
    if (tid < off) {
      r1[tid] += r1[tid + off];
      r2[tid] += r2[tid + off];
      r3[tid] += r3[tid + off];
    }
    __syncthreads();
  }
  if (tid == 0) {
    const float l1 = r1[0] * inv1;
    const float l2 = r2[0] * inv23;
    const float l3 = r3[0] * inv23;
    out[0] = l1 + l2 + l3;
    out[1] = l1;
    out[2] = l2;
    out[3] = l3;
  }
}

extern "C" void kernel_launch(void* const* d_in, const int* in_sizes, int n_in,
                              void* d_out, int out_size, void* d_ws, size_t ws_size,
                              hipStream_t stream) {
  const float* gin = (const float*)d_in[0];
  const float* gtg = (const float*)d_in[1];
  const int nbody = in_sizes[0] / 45;           // [B, 15, 3]
  float* part = (float*)d_ws;                   // NBLK*3 floats = 12 KB

  spike_loss_pass1<<<NBLK, THREADS, 0, stream>>>(gin, gtg, part, nbody);

  const float inv1  = 1.0f / (float)(nbody * 45);   // mean over B*J*C
  const float inv23 = 1.0f / (14.0f * (float)nbody); // / n_conn / B
  spike_loss_pass2<<<1, 256, 0, stream>>>(part, (float*)d_out, NBLK, inv1, inv23);
}